// GCN_65498251264412
// MI455X (gfx1250) — compile-verified
//
#include <hip/hip_runtime.h>
#include <hip/hip_bf16.h>

// ---------------------------------------------------------------------------
// GCN forward for MI455X (gfx1250, wave32).
//   layer: B = act(h) @ W  (fp32 WMMA 16x16x4, W staged K-pair-interleaved in
//          LDS so each B fragment is one aligned ds_load_b64),
//          agg = D^-1/2 (A+I) D^-1/2 B   (self term + per-edge f32 atomics)
//   concat / bias+relu fused into the GEMM A-fragment loads & pooling pass.
// ---------------------------------------------------------------------------

#define N_NODES   200000
#define N_EDGES   640000
#define NUM_GRAPHS   256
#define OUTC          16

typedef __attribute__((ext_vector_type(2))) float v2f;
typedef __attribute__((ext_vector_type(8))) float v8f;

// ---------------- degree / norm ----------------
__global__ void gcn_deg_init(float* deg) {
    int i = blockIdx.x * 256 + threadIdx.x;
    if (i < N_NODES) deg[i] = 1.0f;          // self loop
}

__global__ void gcn_deg_edges(const int* __restrict__ dst, float* deg) {
    int e = blockIdx.x * 256 + threadIdx.x;
    if (e < N_EDGES) atomicAdd(&deg[dst[e]], 1.0f);
}

__global__ void gcn_deg_rsqrt(float* deg) {
    int i = blockIdx.x * 256 + threadIdx.x;
    if (i < N_NODES) deg[i] = rsqrtf(deg[i]);   // in place -> deg_inv_sqrt
}

// ---------------- fp32 WMMA GEMM:  B = act(A) [rows x 128] @ W[128 x 128] --
// MODE 0: A = concat(x[:,0:124], top[:,0:4])          (layer-1 input fusion)
// MODE 1: A = relu(Abuf + bias[col])                  (layer-2 bias/relu fusion)
// LDS W layout: sW[(k>>1)*256 + n*2 + (k&1)]  ->  (K=even,K=odd) pairs are
// adjacent 8B-aligned, so every B fragment is a single ds_load_b64.
template <int MODE>
__global__ __launch_bounds__(256) void gcn_gemm_wmma(
        const float* __restrict__ A0, const float* __restrict__ A1,
        const float* __restrict__ W, float* __restrict__ B, int nRows) {
    __shared__ float sW[128 * 128];              // 64 KB weights in LDS
    for (int i = threadIdx.x; i < 128 * 128; i += 256) {
        int kk = i >> 7, n = i & 127;
        sW[((kk >> 1) << 8) + n * 2 + (kk & 1)] = W[i];
    }
    __syncthreads();

    const int wave = threadIdx.x >> 5;
    const int lane = threadIdx.x & 31;
    const long m0  = ((long)blockIdx.x * 8 + wave) * 16;
    if (m0 >= nRows) return;

    const int half = lane >> 4;                  // 0: K=0,1   1: K=2,3
    const int lr   = lane & 15;
    const int koff = half * 2;
    const long row = m0 + lr;                    // A fragment row for this lane

    auto loadA = [&](int k) -> v2f {             // k is wave-uniform
        v2f r;
        if (MODE == 0) {
            if (k < 124) {
                const float* p = A0 + row * 124 + k + koff;
                r.x = p[0]; r.y = p[1];
            } else {
                const float* p = A1 + row * 4 + (k - 124) + koff;
                r.x = p[0]; r.y = p[1];
            }
        } else {
            const float* p = A0 + row * 128 + k + koff;
            r.x = fmaxf(p[0] + A1[k + koff],     0.0f);
            r.y = fmaxf(p[1] + A1[k + koff + 1], 0.0f);
        }
        return r;
    };

    v8f acc[8];
#pragma unroll
    for (int i = 0; i < 8; ++i) acc[i] = (v8f)(0.0f);

    v2f a = loadA(0);
    for (int k = 0; k < 128; k += 4) {
        v2f an = a;
        if (k + 4 < 128) an = loadA(k + 4);      // prefetch next A fragment
        // base of the (K=k+koff, K=k+koff+1) pair for column lr
        const float* bbase = sW + (((k + koff) >> 1) << 8) + lr * 2;
#pragma unroll
        for (int nt = 0; nt < 8; ++nt) {
            v2f b = *(const v2f*)(bbase + nt * 32);   // one aligned ds_load_b64
            acc[nt] = __builtin_amdgcn_wmma_f32_16x16x4_f32(
                false, a, false, b, (short)0, acc[nt], false, false);
        }
        a = an;
    }
    // D layout: VGPR r -> (M = half*8 + r, N = lr)
#pragma unroll
    for (int nt = 0; nt < 8; ++nt) {
        const int n0 = nt * 16;
#pragma unroll
        for (int r = 0; r < 8; ++r)
            B[(m0 + half * 8 + r) * 128 + n0 + lr] = acc[nt][r];
    }
}

// ---------------- aggregation ----------------
// out[i] = dis[i]^2 * B[i]   (self-loop term, also re-initializes out)
__global__ void gcn_agg_init(const float* __restrict__ B,
                             const float* __restrict__ dis,
                             float* __restrict__ out) {
    long idx = (long)blockIdx.x * 256 + threadIdx.x;
    if (idx >= (long)N_NODES * 128) return;
    int n = (int)(idx >> 7);
    float d = dis[n];
    out[idx] = d * d * B[idx];
}

// one wave per edge: lane handles 4 channels (float4 gather + 4 f32 atomics)
__global__ __launch_bounds__(256) void gcn_agg_edges(
        const float* __restrict__ B, float* __restrict__ out,
        const int* __restrict__ src, const int* __restrict__ dst,
        const float* __restrict__ dis) {
    int e = blockIdx.x * 8 + (threadIdx.x >> 5);
    if (e >= N_EDGES) return;
    const int lane = threadIdx.x & 31;
    const int s = src[e], d = dst[e];
    const float norm = dis[s] * dis[d];
    float4 v = ((const float4*)(B + (long)s * 128))[lane];
    float* op = out + (long)d * 128 + lane * 4;
    atomicAdd(op + 0, norm * v.x);
    atomicAdd(op + 1, norm * v.y);
    atomicAdd(op + 2, norm * v.z);
    atomicAdd(op + 3, norm * v.w);
}

// ---------------- pooling + FC ----------------
__global__ void gcn_pool_zero(float* sums, float* cnts) {
    int i = blockIdx.x * 256 + threadIdx.x;
    if (i < NUM_GRAPHS * 128) sums[i] = 0.0f;
    if (i < NUM_GRAPHS) cnts[i] = 0.0f;
}

// fused relu(h + b2) before mean-pool accumulation
__global__ void gcn_pool_accum(const float* __restrict__ h,
                               const float* __restrict__ bias,
                               const int* __restrict__ batch,
                               float* sums, float* cnts) {
    long idx = (long)blockIdx.x * 256 + threadIdx.x;
    if (idx >= (long)N_NODES * 128) return;
    int n = (int)(idx >> 7), c = (int)(idx & 127);
    int g = batch[n];
    atomicAdd(&sums[g * 128 + c], fmaxf(h[idx] + bias[c], 0.0f));
    if (c == 0) atomicAdd(&cnts[g], 1.0f);
}

__global__ void gcn_fc(const float* __restrict__ sums, const float* __restrict__ cnts,
                       const float* __restrict__ fcW, const float* __restrict__ fcb,
                       float* __restrict__ out) {
    int idx = blockIdx.x * 256 + threadIdx.x;
    if (idx >= NUM_GRAPHS * OUTC) return;
    int g = idx >> 4, o = idx & 15;
    float inv = 1.0f / fmaxf(cnts[g], 1.0f);
    float acc = fcb[o];
    for (int k = 0; k < 128; ++k)
        acc += sums[g * 128 + k] * inv * fcW[k * OUTC + o];
    out[idx] = acc;
}

// ---------------------------------------------------------------------------
extern "C" void kernel_launch(void* const* d_in, const int* in_sizes, int n_in,
                              void* d_out, int out_size, void* d_ws, size_t ws_size,
                              hipStream_t stream) {
    (void)in_sizes; (void)n_in; (void)out_size; (void)ws_size;
    const float* x    = (const float*)d_in[0];
    const float* top  = (const float*)d_in[1];
    const int*   ei   = (const int*)d_in[2];      // [2, E]: src row then dst row
    const int*   bat  = (const int*)d_in[3];
    const float* W1   = (const float*)d_in[4];
    const float* b1   = (const float*)d_in[5];
    const float* W2   = (const float*)d_in[6];
    const float* b2   = (const float*)d_in[7];
    const float* fcW  = (const float*)d_in[8];
    const float* fcb  = (const float*)d_in[9];
    float*       out  = (float*)d_out;

    const int* src = ei;
    const int* dst = ei + N_EDGES;

    char* ws = (char*)d_ws;
    float* A    = (float*)ws; ws += (size_t)N_NODES * 128 * 4;   // agg / h buffer
    float* Bh   = (float*)ws; ws += (size_t)N_NODES * 128 * 4;   // h @ W buffer
    float* dis  = (float*)ws; ws += (size_t)N_NODES * 4;         // deg -> deg^-1/2
    float* sums = (float*)ws; ws += (size_t)NUM_GRAPHS * 128 * 4;
    float* cnts = (float*)ws;

    const int tb = 256;
    const int gN    = (N_NODES + tb - 1) / tb;
    const int gE    = (N_EDGES + tb - 1) / tb;
    const long NC   = (long)N_NODES * 128;
    const int gNC   = (int)((NC + tb - 1) / tb);
    const int gGemm = (N_NODES / 16 + 7) / 8;        // 12500 row-tiles, 8 waves/block
    const int gEdge = (N_EDGES + 7) / 8;             // 1 wave per edge

    // degree / normalization
    gcn_deg_init <<<gN, tb, 0, stream>>>(dis);
    gcn_deg_edges<<<gE, tb, 0, stream>>>(dst, dis);
    gcn_deg_rsqrt<<<gN, tb, 0, stream>>>(dis);

    // layer 1:  Bh = concat(x, top) @ W1   (concat fused into A loads)
    gcn_gemm_wmma<0><<<gGemm, tb, 0, stream>>>(x, top, W1, Bh, N_NODES);
    gcn_agg_init <<<gNC, tb, 0, stream>>>(Bh, dis, A);
    gcn_agg_edges<<<gEdge, tb, 0, stream>>>(Bh, A, src, dst, dis);

    // layer 2:  Bh = relu(A + b1) @ W2     (bias+relu fused into A loads)
    gcn_gemm_wmma<1><<<gGemm, tb, 0, stream>>>(A, b1, W2, Bh, N_NODES);
    gcn_agg_init <<<gNC, tb, 0, stream>>>(Bh, dis, A);
    gcn_agg_edges<<<gEdge, tb, 0, stream>>>(Bh, A, src, dst, dis);

    // mean pool (relu(A + b2) fused) + FC
    gcn_pool_zero <<<(NUM_GRAPHS * 128 + tb - 1) / tb, tb, 0, stream>>>(sums, cnts);
    gcn_pool_accum<<<gNC, tb, 0, stream>>>(A, b2, bat, sums, cnts);
    gcn_fc<<<(NUM_GRAPHS * OUTC + tb - 1) / tb, tb, 0, stream>>>(sums, cnts, fcW, fcb, out);
}